// VolumetricRenderer_62242666054346
// MI455X (gfx1250) — compile-verified
//
#include <hip/hip_runtime.h>
#include <cstdint>

typedef __attribute__((ext_vector_type(2))) float v2f;
typedef __attribute__((ext_vector_type(8))) float v8f;

#define FAR_DELTA 1e10f
#define EPS_T     1e-10f

#define NS    128           // samples per ray
#define TILE  32            // samples per LDS tile
#define NT    (NS / TILE)   // 4 tiles
#define WAVES 2             // waves per block
#define BLOCK (WAVES * 32)
#define RPW   32            // rays per wave (lane == ray)
#define SROW  36            // sigma/z/weights LDS row stride (floats): (36r+s)%64 -> 2-way max
#define CROW  100           // rgb LDS row stride (floats), 16B aligned
#define WAVE_FLOATS (3*RPW*SROW + RPW*CROW + 16*17)   // 3456 + 3200 + 272 = 6928

// ---- CDNA5 async LDS<->global DMA path (ASYNCcnt) via inline asm ----
__device__ __forceinline__ uint32_t lds_lo32(const void* p) {
  // generic LDS pointer: addr[31:0] is the LDS byte offset (ISA 10.2 aperture rules)
  return (uint32_t)(uint64_t)p;
}
__device__ __forceinline__ void async_load_b128(uint32_t lds, const void* g) {
  asm volatile("global_load_async_to_lds_b128 %0, %1, off"
               :: "v"(lds), "v"((uint64_t)g) : "memory");
}
__device__ __forceinline__ void async_load_b32(uint32_t lds, const void* g) {
  asm volatile("global_load_async_to_lds_b32 %0, %1, off"
               :: "v"(lds), "v"((uint64_t)g) : "memory");
}
__device__ __forceinline__ void async_store_b128(void* g, uint32_t lds) {
  asm volatile("global_store_async_from_lds_b128 %0, %1, off"
               :: "v"((uint64_t)g), "v"(lds) : "memory");
}
__device__ __forceinline__ void wait_asynccnt0() {
  asm volatile("s_wait_asynccnt 0x0" ::: "memory");
}
__device__ __forceinline__ void wait_dscnt0() {
  asm volatile("s_wait_dscnt 0x0" ::: "memory");
}

__global__ __launch_bounds__(BLOCK)
void VolumetricRenderer_62242666054346_kernel(
    const float* __restrict__ rgb,    // [N,128,3]
    const float* __restrict__ sigma,  // [N,128]
    const float* __restrict__ zvals,  // [N,128]
    const unsigned char* __restrict__ whiteFlag,
    float* __restrict__ compOut,      // [N,3]
    float* __restrict__ depthOut,     // [N]
    float* __restrict__ opacOut,      // [N]
    float* __restrict__ wOut,         // [N,128]
    int nray)
{
  __shared__ __align__(16) float smem[WAVES * WAVE_FLOATS];

  const int tid  = threadIdx.x;
  const int wave = tid >> 5;
  const int lane = tid & 31;
  const int rayBase = blockIdx.x * (WAVES * RPW) + wave * RPW;
  if (rayBase >= nray) return;

  float* Ssig = smem + wave * WAVE_FLOATS;
  float* Sz   = Ssig + RPW * SROW;
  float* Sc   = Sz   + RPW * SROW;   // rgb tile, [ray][s*3+c]
  float* Sw   = Sc   + RPW * CROW;   // weights tile
  float* Sred = Sw   + RPW * SROW;   // 16x17 fragment transpose scratch

  float T = 1.0f;                    // exclusive transmittance (cumprod carry)
  float depthAcc = 0.f, cr = 0.f, cg = 0.f, cb = 0.f;
  v8f acc0 = {}; v8f acc1 = {};      // WMMA opacity accumulators (rays 0-15 / 16-31)
  v2f ones; ones.x = 1.f; ones.y = 1.f;

  for (int t = 0; t < NT; ++t) {
    // ---- issue coalesced async loads: 16B segments, consecutive lanes contiguous ----
    for (int i = 0; i < 8; ++i) {                 // sigma & z: 8 segs/ray * 32 rays
      int f = i * 32 + lane;
      int ray = f >> 3, seg = f & 7;
      const float* gs = sigma + (size_t)(rayBase + ray) * NS + t * TILE + seg * 4;
      const float* gz = zvals + (size_t)(rayBase + ray) * NS + t * TILE + seg * 4;
      async_load_b128(lds_lo32(&Ssig[ray * SROW + seg * 4]), gs);
      async_load_b128(lds_lo32(&Sz  [ray * SROW + seg * 4]), gz);
    }
    if (t < NT - 1) {                             // z overlap column for last delta
      const float* gz = zvals + (size_t)(rayBase + lane) * NS + t * TILE + TILE;
      async_load_b32(lds_lo32(&Sz[lane * SROW + TILE]), gz);
    }
    for (int i = 0; i < 24; ++i) {                // rgb: 24 segs/ray * 32 rays
      int f = i * 32 + lane;
      int ray = f / 24, seg = f % 24;
      const float* gc = rgb + (size_t)(rayBase + ray) * (NS * 3) + t * (TILE * 3) + seg * 4;
      async_load_b128(lds_lo32(&Sc[ray * CROW + seg * 4]), gc);
    }
    wait_asynccnt0();   // tile loaded; also drains previous tile's weight store (Sw reuse)

    // ---- per-lane (lane == ray) sequential transmittance scan over this tile ----
    const int r = lane;
    for (int s = 0; s < TILE; ++s) {
      float sg = Ssig[r * SROW + s];
      float zc = Sz  [r * SROW + s];
      float zn = Sz  [r * SROW + s + 1];           // overlap col makes this valid for t<3
      bool last = (t == NT - 1) && (s == TILE - 1);
      float delta = last ? FAR_DELTA : (zn - zc);
      float dens  = fmaxf(sg, 0.f);                // relu
      float alpha = 1.f - __expf(-delta * dens);
      float w     = alpha * T;
      depthAcc += w * zc;
      cr += w * Sc[r * CROW + s * 3 + 0];
      cg += w * Sc[r * CROW + s * 3 + 1];
      cb += w * Sc[r * CROW + s * 3 + 2];
      Sw[r * SROW + s] = w;
      T *= (1.f - alpha + EPS_T);
    }

    // ---- opacity via matrix row-sum idiom: D += W(16x4) x ones(4x16), col 0 = sum_k w ----
    {
      int m    = lane & 15;                        // A layout: lanes 0-15 K0/K1, 16-31 K2/K3
      int koff = (lane >> 4) * 2;
      for (int kk = 0; kk < TILE / 4; ++kk) {
        v2f a0, a1;
        a0.x = Sw[(m     ) * SROW + kk * 4 + koff];
        a0.y = Sw[(m     ) * SROW + kk * 4 + koff + 1];
        a1.x = Sw[(m + 16) * SROW + kk * 4 + koff];
        a1.y = Sw[(m + 16) * SROW + kk * 4 + koff + 1];
        acc0 = __builtin_amdgcn_wmma_f32_16x16x4_f32(false, a0, false, ones,
                                                     (short)0, acc0, false, false);
        acc1 = __builtin_amdgcn_wmma_f32_16x16x4_f32(false, a1, false, ones,
                                                     (short)0, acc1, false, false);
      }
    }

    wait_dscnt0();      // weight tile committed to LDS before the DMA store reads it
    for (int i = 0; i < 8; ++i) {                  // coalesced async store of weights
      int f = i * 32 + lane;
      int ray = f >> 3, seg = f & 7;
      float* gw = wOut + (size_t)(rayBase + ray) * NS + t * TILE + seg * 4;
      async_store_b128(gw, lds_lo32(&Sw[ray * SROW + seg * 4]));
    }
    // s_endpgm's implicit wait-idle drains the final stores
  }

  // ---- extract per-ray opacity = D[m][0] via 16x17 LDS transpose of the C/D fragment ----
  int n  = lane & 15;
  int mb = (lane >> 4) * 8;                        // C/D layout: VGPR j -> M=j / M=j+8
  for (int j = 0; j < 8; ++j) Sred[(mb + j) * 17 + n] = acc0[j];
  wait_dscnt0();
  float o0 = Sred[(lane & 15) * 17 + 0];
  for (int j = 0; j < 8; ++j) Sred[(mb + j) * 17 + n] = acc1[j];
  wait_dscnt0();
  float o1 = Sred[(lane & 15) * 17 + 0];
  float opacity = (lane < 16) ? o0 : o1;

  float whiteAdd = (*whiteFlag != 0) ? (1.f - opacity) : 0.f;
  int ray = rayBase + lane;
  opacOut [ray] = opacity;
  depthOut[ray] = depthAcc;
  compOut[ray * 3 + 0] = cr + whiteAdd;
  compOut[ray * 3 + 1] = cg + whiteAdd;
  compOut[ray * 3 + 2] = cb + whiteAdd;
}

extern "C" void kernel_launch(void* const* d_in, const int* in_sizes, int n_in,
                              void* d_out, int out_size, void* d_ws, size_t ws_size,
                              hipStream_t stream) {
  const float* rgb   = (const float*)d_in[0];
  const float* sigma = (const float*)d_in[1];
  const float* zv    = (const float*)d_in[2];
  const unsigned char* wflag = (const unsigned char*)d_in[3];

  int nray = in_sizes[1] / NS;                 // sigma is [N,128]
  float* out   = (float*)d_out;                // [comp(3N) | depth(N) | opac(N) | weights(128N)]
  float* comp  = out;
  float* depth = out + (size_t)3 * nray;
  float* opac  = out + (size_t)4 * nray;
  float* wts   = out + (size_t)5 * nray;

  int blocks = (nray + WAVES * RPW - 1) / (WAVES * RPW);
  hipLaunchKernelGGL(VolumetricRenderer_62242666054346_kernel,
                     dim3(blocks), dim3(BLOCK), 0, stream,
                     rgb, sigma, zv, wflag, comp, depth, opac, wts, nray);
}